// AdditiveAttention_11596411699917
// MI455X (gfx1250) — compile-verified
//
#include <hip/hip_runtime.h>
#include <hip/hip_bf16.h>
#include <math.h>

// Problem sizes (fixed by the reference)
#define BATCH 16
#define QLEN  128
#define KLEN  512
#define HDIM  128
#define DDIM  128   // DQ == DK == DV == 128
#define MASK_VALUE (-1000000.0f)

typedef __attribute__((ext_vector_type(2))) float v2f;
typedef __attribute__((ext_vector_type(8))) float v8f;

// D = A(16x4) * B(4x16) + C(16x16), full fp32 via V_WMMA_F32_16X16X4_F32.
// 8-arg pattern: (neg_a, A, neg_b, B, c_mod, C, reuse_a, reuse_b)
__device__ __forceinline__ v8f wmma_f32_4(v2f a, v2f b, v8f c) {
    return __builtin_amdgcn_wmma_f32_16x16x4_f32(false, a, false, b,
                                                 (short)0, c, false, false);
}

// ---------------------------------------------------------------------------
// Generic row-major GEMM  C[rows x 128] = A[rows x 128] * W[128 x 128]
// One wave (32 lanes) computes one 16x16 tile of C; 8 waves per block.
// Grid must be sized so every wave maps to a valid tile (rows % 16 == 0).
// ---------------------------------------------------------------------------
__global__ void proj_gemm_wmma(const float* __restrict__ A,
                               const float* __restrict__ W,
                               float* __restrict__ C) {
    const int lane   = threadIdx.x & 31;
    const int wave   = (blockIdx.x * blockDim.x + threadIdx.x) >> 5;
    const int tilesN = HDIM / 16;            // 8
    const int row0   = (wave / tilesN) * 16;
    const int col0   = (wave % tilesN) * 16;

    const int lm = lane & 15;                // M (for A) / N (for B) index
    const int hi = lane >> 4;                // lane-half selects K pair

    v8f acc = {};
    #pragma unroll 4
    for (int kd = 0; kd < DDIM; kd += 4) {
        const int ka = kd + 2 * hi;          // this half-wave's K base
        v2f a, b;
        // A 16x4 f32 layout: lanes 0-15 -> M=lane, VGPR0/1 = K ka, ka+1
        a.x = A[(row0 + lm) * DDIM + ka];
        a.y = A[(row0 + lm) * DDIM + ka + 1];
        // B 4x16 f32 layout (mirror of A with N on lanes)
        b.x = W[(ka)     * HDIM + col0 + lm];
        b.y = W[(ka + 1) * HDIM + col0 + lm];
        acc = wmma_f32_4(a, b, acc);
    }
    // C/D 16x16 f32 layout: VGPR r, lanes 0-15 -> M=r, lanes 16-31 -> M=r+8
    #pragma unroll
    for (int r = 0; r < 8; ++r)
        C[(row0 + r + 8 * hi) * HDIM + col0 + lm] = acc[r];
}

// ---------------------------------------------------------------------------
// Fused additive-attention scores + mask + softmax.
// One 512-thread block per (b, q); thread k owns key k.
//   score[k] = sum_h w_v[h] * tanh(qp[b,q,h] + kp[b,k,h])   (tanh -> v_tanh_f32)
// Never materializes the (B,Q,K,H) features tensor (536 MB in the reference).
// ---------------------------------------------------------------------------
__global__ void score_softmax(const float* __restrict__ qp,
                              const float* __restrict__ kp,
                              const float* __restrict__ wv,
                              const int*   __restrict__ valid_lens,
                              float* __restrict__ attn) {
    const int bq = blockIdx.x;
    const int b  = bq >> 7;          // / QLEN
    const int q  = bq & (QLEN - 1);
    const int k  = threadIdx.x;      // 0..511

    __shared__ float qrow[HDIM];
    __shared__ float wvs[HDIM];
    __shared__ float red[KLEN];

    if (k < HDIM) {
        qrow[k] = qp[(b * QLEN + q) * HDIM + k];
        wvs[k]  = wv[k];
    }
    __syncthreads();

    const float4* krow = (const float4*)(kp + (size_t)(b * KLEN + k) * HDIM);
    float s = 0.0f;
    #pragma unroll 4
    for (int h4 = 0; h4 < HDIM / 4; ++h4) {
        float4 kv = krow[h4];
        const int h = h4 * 4;
        s += wvs[h + 0] * tanhf(qrow[h + 0] + kv.x);
        s += wvs[h + 1] * tanhf(qrow[h + 1] + kv.y);
        s += wvs[h + 2] * tanhf(qrow[h + 2] + kv.z);
        s += wvs[h + 3] * tanhf(qrow[h + 3] + kv.w);
    }

    if (k >= valid_lens[b]) s = MASK_VALUE;

    // block softmax over 512 keys: max, then sum of exp
    red[k] = s;
    __syncthreads();
    for (int off = KLEN / 2; off > 0; off >>= 1) {
        if (k < off) red[k] = fmaxf(red[k], red[k + off]);
        __syncthreads();
    }
    const float m = red[0];
    __syncthreads();

    const float e = expf(s - m);
    red[k] = e;
    __syncthreads();
    for (int off = KLEN / 2; off > 0; off >>= 1) {
        if (k < off) red[k] += red[k + off];
        __syncthreads();
    }
    const float inv = 1.0f / red[0];

    attn[((size_t)(b * QLEN + q)) * KLEN + k] = e * inv;
}

// ---------------------------------------------------------------------------
// out[b] = attn[b] (128x512) * values[b] (512x128), fp32 WMMA.
// One wave per 16x16 output tile; 64 tiles per batch, 1024 waves total.
// ---------------------------------------------------------------------------
__global__ void out_gemm_wmma(const float* __restrict__ attn,
                              const float* __restrict__ vals,
                              float* __restrict__ out) {
    const int lane  = threadIdx.x & 31;
    const int gwave = (blockIdx.x * blockDim.x + threadIdx.x) >> 5;
    const int tilesN   = DDIM / 16;          // 8
    const int tilesPerB = (QLEN / 16) * tilesN; // 64
    const int b    = gwave / tilesPerB;
    const int t    = gwave % tilesPerB;
    const int row0 = (t / tilesN) * 16;
    const int col0 = (t % tilesN) * 16;

    const float* Ab = attn + (size_t)b * QLEN * KLEN;
    const float* Bb = vals + (size_t)b * KLEN * DDIM;
    float*       Cb = out  + (size_t)b * QLEN * DDIM;

    const int lm = lane & 15;
    const int hi = lane >> 4;

    v8f acc = {};
    #pragma unroll 4
    for (int kd = 0; kd < KLEN; kd += 4) {
        const int ka = kd + 2 * hi;
        v2f a, bb;
        a.x  = Ab[(row0 + lm) * KLEN + ka];
        a.y  = Ab[(row0 + lm) * KLEN + ka + 1];
        bb.x = Bb[(ka)     * DDIM + col0 + lm];
        bb.y = Bb[(ka + 1) * DDIM + col0 + lm];
        acc = wmma_f32_4(a, bb, acc);
    }
    #pragma unroll
    for (int r = 0; r < 8; ++r)
        Cb[(row0 + r + 8 * hi) * DDIM + col0 + lm] = acc[r];
}

// ---------------------------------------------------------------------------
extern "C" void kernel_launch(void* const* d_in, const int* in_sizes, int n_in,
                              void* d_out, int out_size, void* d_ws, size_t ws_size,
                              hipStream_t stream) {
    const float* queries   = (const float*)d_in[0]; // (B,Q,128)
    const float* keys      = (const float*)d_in[1]; // (B,K,128)
    const float* values    = (const float*)d_in[2]; // (B,K,128)
    const int*   valid     = (const int*)  d_in[3]; // (B,)
    const float* W_q       = (const float*)d_in[4]; // (128,128)
    const float* W_k       = (const float*)d_in[5]; // (128,128)
    const float* w_v       = (const float*)d_in[6]; // (128,)
    float*       out       = (float*)d_out;         // (B,Q,128)

    // Workspace layout (floats): qp | kp | attn  = 1MB + 4MB + 4MB
    float* qp   = (float*)d_ws;
    float* kp   = qp + (size_t)BATCH * QLEN * HDIM;
    float* attn = kp + (size_t)BATCH * KLEN * HDIM;

    // q-projection: 2048 rows -> 128*8 = 1024 tiles -> 1024 waves / 8 = 128 blocks
    proj_gemm_wmma<<<128, 256, 0, stream>>>(queries, W_q, qp);
    // k-projection: 8192 rows -> 512*8 = 4096 tiles -> 512 blocks
    proj_gemm_wmma<<<512, 256, 0, stream>>>(keys, W_k, kp);
    // fused scores + mask + softmax: one block per (b,q)
    score_softmax<<<BATCH * QLEN, KLEN, 0, stream>>>(qp, kp, w_v, valid, attn);
    // attn @ values: 16 batches * 64 tiles = 1024 waves -> 128 blocks
    out_gemm_wmma<<<128, 256, 0, stream>>>(attn, values, out);
}